// GetDensity_38242388803782
// MI455X (gfx1250) — compile-verified
//
#include <hip/hip_runtime.h>
#include <math.h>

#define NTYPE 4
#define NWAVE 8
#define NORBIT 128
#define NEIGH 64
#define NB 32
#define NATOM 256
#define NPAIR_PER (NATOM * NEIGH)
#define NPAIR_TOT (NB * NPAIR_PER)     // 524288
#define TOTNATOM (NB * NATOM)          // 8192
#define NPARA 13
#define PI_OVER_CUTOFF 0.62831853071795864769f

typedef float v2f __attribute__((ext_vector_type(2)));
typedef float v8f __attribute__((ext_vector_type(8)));

__device__ __forceinline__ v8f wmma_f32_16x16x4(v2f a, v2f b, v8f c) {
  // 8 args: (neg_a, A, neg_b, B, c_mod, C, reuse_a, reuse_b)
  return __builtin_amdgcn_wmma_f32_16x16x4_f32(false, a, false, b, (short)0, c,
                                               false, false);
}

// ---------------------------------------------------------------- init / zero
__global__ void k_init_orb(const float* __restrict__ params,
                           const int* __restrict__ species,
                           float* __restrict__ orb) {
  int t = blockIdx.x * blockDim.x + threadIdx.x;   // over TOTNATOM*NWAVE
  int a = t / NWAVE, w = t % NWAVE;
  orb[t] = params[species[a] * NWAVE + w];
}

__global__ void k_zero(float* __restrict__ p, int n) {
  int t = blockIdx.x * blockDim.x + threadIdx.x;
  if (t < n) p[t] = 0.0f;
}

// ------------------------------------------------- pair stage + scatter (B1)
// Recomputed each obtain: geometry, cutoff, radial, angular; scatter-add the
// weighted 13x8 orbital into sum_worbital[center][13][8] with f32 HW atomics.
__global__ void k_pairs(const float* __restrict__ cart,
                        const float* __restrict__ shifts,
                        const float* __restrict__ rs,
                        const float* __restrict__ inta,
                        const int* __restrict__ species,
                        const int* __restrict__ atom_index,
                        const float* __restrict__ orb,
                        float* __restrict__ sum_w) {
  int p = blockIdx.x * blockDim.x + threadIdx.x;   // over NPAIR_TOT (exact)
  int b = p / NPAIR_PER;
  int gi = b * NATOM + atom_index[p];               // center (plane 0)
  int gj = b * NATOM + atom_index[NPAIR_TOT + p];   // neighbor (plane 1)

  float dx = cart[gi * 3 + 0] - cart[gj * 3 + 0] + shifts[p * 3 + 0];
  float dy = cart[gi * 3 + 1] - cart[gj * 3 + 1] + shifts[p * 3 + 1];
  float dz = cart[gi * 3 + 2] - cart[gj * 3 + 2] + shifts[p * 3 + 2];
  float d = sqrtf(dx * dx + dy * dy + dz * dz);
  float fc = 0.5f * cosf(d * PI_OVER_CUTOFF) + 0.5f;
  fc = fc * fc;

  int sp = species[gj];
  float wv[NWAVE];
#pragma unroll
  for (int w = 0; w < NWAVE; ++w) {
    float t = d - rs[sp * NWAVE + w];
    wv[w] = expf(inta[sp * NWAVE + w] * t * t) * orb[gj * NWAVE + w];
  }

  float ang[NPARA];
  float dv[3] = {dx, dy, dz};
  ang[0] = fc;
#pragma unroll
  for (int a2 = 0; a2 < 3; ++a2) ang[1 + a2] = fc * dv[a2];
#pragma unroll
  for (int a2 = 0; a2 < 3; ++a2)
#pragma unroll
    for (int b2 = 0; b2 < 3; ++b2) ang[4 + a2 * 3 + b2] = ang[1 + a2] * dv[b2];

  float* dst = sum_w + gi * (NPARA * NWAVE);
#pragma unroll
  for (int j = 0; j < NPARA; ++j)
#pragma unroll
    for (int w = 0; w < NWAVE; ++w)
      unsafeAtomicAdd(dst + j * NWAVE + w, ang[j] * wv[w]);
}

// ------------------------------------- einsum + square-sum via fp32 WMMA (B2)
// One wave32 per 16-atom tile. For each 16-col N-tile: for ip in {0,1,2} load
// B once; for each para of that ip, A(16x8) @ B(8x128-tile) via two K=4 WMMAs,
// square D, accumulate over the 13 paras. K layout per ISA: lanes 0-15 hold
// K={0,1} in the 2 A/B VGPRs, lanes 16-31 hold K={2,3}.
__global__ void k_density(const float* __restrict__ sum_w,
                          const float* __restrict__ hyper,
                          float* __restrict__ density) {
  int wave = threadIdx.x >> 5;
  int lane = threadIdx.x & 31;
  int tile = blockIdx.x * 8 + wave;   // 0..511
  int lrow = lane & 15;
  int lhi = lane >> 4;                // 0 or 1
  int khalf = lhi * 2;

  for (int nt = 0; nt < 8; ++nt) {
    int n = nt * 16 + lrow;
    v8f acc = {};
    int para = 0;
    for (int ip = 0; ip < 3; ++ip) {
      const float* hb = hyper + ip * (NWAVE * NORBIT);
      v2f Blo, Bhi;
      Blo.x = hb[(khalf + 0) * NORBIT + n];
      Blo.y = hb[(khalf + 1) * NORBIT + n];
      Bhi.x = hb[(khalf + 4) * NORBIT + n];
      Bhi.y = hb[(khalf + 5) * NORBIT + n];
      int npj = (ip == 0) ? 1 : ((ip == 1) ? 3 : 9);
      for (int q = 0; q < npj; ++q, ++para) {
        const float* ab = sum_w + ((tile * 16 + lrow) * NPARA + para) * NWAVE;
        v2f Alo = {ab[khalf + 0], ab[khalf + 1]};
        v2f Ahi = {ab[khalf + 4], ab[khalf + 5]};
        v8f c = {};
        c = wmma_f32_16x16x4(Alo, Blo, c);
        c = wmma_f32_16x16x4(Ahi, Bhi, c);
#pragma unroll
        for (int r = 0; r < 8; ++r) acc[r] += c[r] * c[r];
      }
    }
#pragma unroll
    for (int r = 0; r < 8; ++r) {
      int atom = tile * 16 + r + lhi * 8;   // C/D layout: VGPR r -> M=r (+8)
      density[atom * NORBIT + nt * 16 + lrow] = acc[r];
    }
  }
}

// ------------------------------- orb_coeff += tanh(density @ W + e)  (WMMA)
// 8192x128 @ 128x8, N padded to 16 with zero columns; 32 chained K=4 WMMAs.
__global__ void k_update(const float* __restrict__ density,
                         const float* __restrict__ oc_w,
                         const float* __restrict__ oc_e,
                         const int* __restrict__ species,
                         float* __restrict__ orb) {
  int wave = threadIdx.x >> 5;
  int lane = threadIdx.x & 31;
  int tile = blockIdx.x * 8 + wave;
  int lrow = lane & 15;
  int lhi = lane >> 4;

  v8f acc = {};
  const float* arow = density + (tile * 16 + lrow) * NORBIT;
  bool valid = (lrow < NWAVE);
#pragma unroll 4
  for (int kb = 0; kb < 32; ++kb) {
    int k = kb * 4 + lhi * 2;
    v2f A = {arow[k], arow[k + 1]};
    v2f B;
    B.x = valid ? oc_w[(k + 0) * NWAVE + lrow] : 0.0f;
    B.y = valid ? oc_w[(k + 1) * NWAVE + lrow] : 0.0f;
    acc = wmma_f32_16x16x4(A, B, acc);
  }
  if (valid) {
#pragma unroll
    for (int r = 0; r < 8; ++r) {
      int atom = tile * 16 + r + lhi * 8;
      float v = tanhf(acc[r] + oc_e[species[atom] * NWAVE + lrow]);
      orb[atom * NWAVE + lrow] += v;
    }
  }
}

// ---------------------------------------------------------------------------
extern "C" void kernel_launch(void* const* d_in, const int* in_sizes, int n_in,
                              void* d_out, int out_size, void* d_ws,
                              size_t ws_size, hipStream_t stream) {
  const float* cart = (const float*)d_in[0];
  const float* shifts = (const float*)d_in[1];
  const float* rs = (const float*)d_in[2];
  const float* inta = (const float*)d_in[3];
  const float* params = (const float*)d_in[4];
  const float* hyper = (const float*)d_in[5];
  const float* oc_w0 = (const float*)d_in[6];
  const float* oc_e0 = (const float*)d_in[7];
  const float* oc_w1 = (const float*)d_in[8];
  const float* oc_e1 = (const float*)d_in[9];
  const int* species = (const int*)d_in[11];
  const int* atom_index = (const int*)d_in[12];

  float* density = (float*)d_out;                    // 8192 x 128
  float* orb = (float*)d_ws;                         // 8192 x 8
  float* sum_w = orb + TOTNATOM * NWAVE;             // 8192 x 13 x 8

  const int SW_N = TOTNATOM * NPARA * NWAVE;
  const float* Ws[2] = {oc_w0, oc_w1};
  const float* Es[2] = {oc_e0, oc_e1};

  k_init_orb<<<(TOTNATOM * NWAVE) / 256, 256, 0, stream>>>(params, species, orb);
  for (int it = 0; it < 3; ++it) {
    k_zero<<<(SW_N + 255) / 256, 256, 0, stream>>>(sum_w, SW_N);
    k_pairs<<<NPAIR_TOT / 256, 256, 0, stream>>>(cart, shifts, rs, inta, species,
                                                 atom_index, orb, sum_w);
    k_density<<<TOTNATOM / 16 / 8, 256, 0, stream>>>(sum_w, hyper, density);
    if (it < 2)
      k_update<<<TOTNATOM / 16 / 8, 256, 0, stream>>>(density, Ws[it], Es[it],
                                                      species, orb);
  }
}